// HeteroGNN_33303176413369
// MI455X (gfx1250) — compile-verified
//
#include <hip/hip_runtime.h>
#include <hip/hip_bf16.h>

typedef float v2f __attribute__((ext_vector_type(2)));
typedef float v8f __attribute__((ext_vector_type(8)));

#define DFEAT 128   // input feature dim
#define HFEAT 128   // hidden dim
#define NOUT  5     // classifier outputs

// ---------------------------------------------------------------------------
// Phase 1: edge scatter (mean aggregation numerators + counts), both directions.
// One thread per (edge, 4-float chunk): float4 gather + 4 f32 atomics per dir.
// Accumulators are L2-resident (14 MB total) -> atomics resolve in L2.
// ---------------------------------------------------------------------------
__global__ void hgnn_scatter_kernel(const float* __restrict__ x_drug,
                                    const float* __restrict__ x_prot,
                                    const int*   __restrict__ src_drug,
                                    const int*   __restrict__ dst_prot,
                                    float* __restrict__ sum_p, float* __restrict__ cnt_p,
                                    float* __restrict__ sum_d, float* __restrict__ cnt_d,
                                    int E) {
  int tid = blockIdx.x * blockDim.x + threadIdx.x;
  if (tid >= E * (DFEAT / 4)) return;
  int e = tid >> 5;          // DFEAT/4 == 32 chunks per edge
  int c = (tid & 31) << 2;   // feature offset (multiple of 4)
  int s = src_drug[e];
  int d = dst_prot[e];

  // drug -> protein
  const float4 vd = *(const float4*)(x_drug + (size_t)s * DFEAT + c);
  float* p = sum_p + (size_t)d * DFEAT + c;
  atomicAdd(p + 0, vd.x); atomicAdd(p + 1, vd.y);
  atomicAdd(p + 2, vd.z); atomicAdd(p + 3, vd.w);

  // protein -> drug (reverse edges)
  const float4 vp = *(const float4*)(x_prot + (size_t)d * DFEAT + c);
  float* q = sum_d + (size_t)s * DFEAT + c;
  atomicAdd(q + 0, vp.x); atomicAdd(q + 1, vp.y);
  atomicAdd(q + 2, vp.z); atomicAdd(q + 3, vp.w);

  if (c == 0) {
    atomicAdd(cnt_p + d, 1.0f);
    atomicAdd(cnt_d + s, 1.0f);
  }
}

// ---------------------------------------------------------------------------
// Phase 2: SAGEConv GEMM via V_WMMA_F32_16X16X4_F32 (exact fp32).
//   Hout = (sumA * 1/max(cnt,1)) @ Wl + bl + X @ Wr        [M x HFEAT]
// Block = 256 threads = 8 wave32; blockIdx.x selects a 16-row stripe and the
// 8 waves cover the 8 N-tiles (HFEAT/16 == 8), sharing the same A rows via
// WGP-local cache. 64 WMMA issues per wave (32 mean-term + 32 root-term).
// Fragment layouts per CDNA5 ISA 7.12.2:
//   A 16x4 f32 : lane&15 = M row, lane half -> K pair {0,1} vs {2,3}
//   B  4x16 f32: lane&15 = N col, lane half -> K pair {0,1} vs {2,3}
//   C/D        : VGPR r -> row r (lanes 0-15) / r+8 (lanes 16-31), N = lane&15
// ---------------------------------------------------------------------------
__global__ void hgnn_sage_gemm_wmma(const float* __restrict__ sumA,
                                    const float* __restrict__ cnt,
                                    const float* __restrict__ X,
                                    const float* __restrict__ Wl,
                                    const float* __restrict__ bl,
                                    const float* __restrict__ Wr,
                                    float* __restrict__ Hout,
                                    int M) {
  const int lane = threadIdx.x & 31;
  const int half = lane >> 4;            // 0: lanes 0-15, 1: lanes 16-31
  const int l15  = lane & 15;
  const int tm   = blockIdx.x;           // 16-row stripe
  const int tn   = threadIdx.x >> 5;     // wave id == N-tile (0..7)

  int row = tm * 16 + l15;
  if (row >= M) row = M - 1;             // tail clamp (reads only; stores guarded)
  const int n    = tn * 16 + l15;
  const int koff = half * 2;

  const float inv = 1.0f / fmaxf(cnt[row], 1.0f);  // mean normalization, per-lane const
  const float* sr = sumA + (size_t)row * DFEAT;
  const float* xr = X    + (size_t)row * DFEAT;

  v8f acc = {};
  for (int k = 0; k < DFEAT; k += 4) {
    v2f a1, b1, a2, b2;
    a1[0] = sr[k + koff] * inv;
    a1[1] = sr[k + koff + 1] * inv;
    b1[0] = Wl[(k + koff) * HFEAT + n];
    b1[1] = Wl[(k + koff + 1) * HFEAT + n];
    acc = __builtin_amdgcn_wmma_f32_16x16x4_f32(false, a1, false, b1,
                                                (short)0, acc, false, false);
    a2[0] = xr[k + koff];
    a2[1] = xr[k + koff + 1];
    b2[0] = Wr[(k + koff) * HFEAT + n];
    b2[1] = Wr[(k + koff + 1) * HFEAT + n];
    acc = __builtin_amdgcn_wmma_f32_16x16x4_f32(false, a2, false, b2,
                                                (short)0, acc, false, false);
  }

  const float bn = bl[n];
  float* outp = Hout + (size_t)(tm * 16) * HFEAT + tn * 16 + l15;
  if (tm * 16 + 16 <= M) {
    // full stripe: wave-uniform fast path, branch-free stores
    #pragma unroll
    for (int r = 0; r < 8; ++r) {
      outp[(size_t)(r + half * 8) * HFEAT] = acc[r] + bn;
    }
  } else {
    // partial tail stripe (not hit for M % 16 == 0)
    #pragma unroll
    for (int r = 0; r < 8; ++r) {
      const int m = tm * 16 + r + half * 8;
      if (m < M) Hout[(size_t)m * HFEAT + tn * 16 + l15] = acc[r] + bn;
    }
  }
}

// ---------------------------------------------------------------------------
// Phase 3: edge classifier. out[e] = concat(h_drug[el_d[e]], h_prot[el_p[e]]) @ W_lin + b
// One wave per edge: lanes 0-15 cover the h_drug half (k=0..127, 8 floats each),
// lanes 16-31 the h_prot half (k=128..255). W_lin (2H x 5 = 5KB) cached in LDS.
// Butterfly reduce 5 accumulators over 32 lanes (wave32-correct).
// ---------------------------------------------------------------------------
__global__ void hgnn_edge_mlp_kernel(const float* __restrict__ h_drug,
                                     const float* __restrict__ h_prot,
                                     const int*   __restrict__ el_drug,
                                     const int*   __restrict__ el_prot,
                                     const float* __restrict__ W_lin,
                                     const float* __restrict__ b_lin,
                                     float* __restrict__ out,
                                     int L) {
  __shared__ float sW[2 * HFEAT * NOUT];
  __shared__ float sb[NOUT];
  for (int i = threadIdx.x; i < 2 * HFEAT * NOUT; i += blockDim.x) sW[i] = W_lin[i];
  if (threadIdx.x < NOUT) sb[threadIdx.x] = b_lin[threadIdx.x];
  __syncthreads();

  const int lane = threadIdx.x & 31;
  const int wave = threadIdx.x >> 5;
  const int wavesPerBlock = blockDim.x >> 5;
  const int e = blockIdx.x * wavesPerBlock + wave;
  if (e >= L) return;

  const int rd = el_drug[e];
  const int rp = el_prot[e];
  const float* rowp = (lane < 16)
      ? (h_drug + (size_t)rd * HFEAT + lane * 8)
      : (h_prot + (size_t)rp * HFEAT + (lane - 16) * 8);
  const float4 v0 = *(const float4*)(rowp + 0);
  const float4 v1 = *(const float4*)(rowp + 4);
  const float x[8] = { v0.x, v0.y, v0.z, v0.w, v1.x, v1.y, v1.z, v1.w };

  const int kbase = lane * 8;   // global k index in [0, 256)
  float acc[NOUT] = {0.f, 0.f, 0.f, 0.f, 0.f};
  #pragma unroll
  for (int j = 0; j < 8; ++j) {
    const float* wrow = sW + (kbase + j) * NOUT;
    #pragma unroll
    for (int o = 0; o < NOUT; ++o) acc[o] += x[j] * wrow[o];
  }

  #pragma unroll
  for (int m = 16; m > 0; m >>= 1) {
    #pragma unroll
    for (int o = 0; o < NOUT; ++o) acc[o] += __shfl_xor(acc[o], m, 32);
  }

  if (lane == 0) {
    #pragma unroll
    for (int o = 0; o < NOUT; ++o) out[(size_t)e * NOUT + o] = acc[o] + sb[o];
  }
}

// ---------------------------------------------------------------------------
extern "C" void kernel_launch(void* const* d_in, const int* in_sizes, int n_in,
                              void* d_out, int out_size, void* d_ws, size_t ws_size,
                              hipStream_t stream) {
  const float* x_drug = (const float*)d_in[0];
  const float* x_prot = (const float*)d_in[1];
  const float* Wl_dp  = (const float*)d_in[2];
  const float* bl_dp  = (const float*)d_in[3];
  const float* Wr_dp  = (const float*)d_in[4];
  const float* Wl_pd  = (const float*)d_in[5];
  const float* bl_pd  = (const float*)d_in[6];
  const float* Wr_pd  = (const float*)d_in[7];
  const float* W_lin  = (const float*)d_in[8];
  const float* b_lin  = (const float*)d_in[9];
  const int* src_drug = (const int*)d_in[10];
  const int* dst_prot = (const int*)d_in[11];
  const int* el_drug  = (const int*)d_in[12];
  const int* el_prot  = (const int*)d_in[13];

  const int n_drug = in_sizes[0] / DFEAT;
  const int n_prot = in_sizes[1] / DFEAT;
  const int E = in_sizes[10];
  const int L = in_sizes[12];

  // Workspace carve-out (256B aligned): accumulators first (zeroed), then h buffers.
  char* ws = (char*)d_ws;
  size_t off = 0;
  auto carve = [&](size_t bytes) -> void* {
    void* p = ws + off;
    off += (bytes + 255) & ~(size_t)255;
    return p;
  };
  float* sum_p = (float*)carve((size_t)n_prot * DFEAT * sizeof(float));
  float* cnt_p = (float*)carve((size_t)n_prot * sizeof(float));
  float* sum_d = (float*)carve((size_t)n_drug * DFEAT * sizeof(float));
  float* cnt_d = (float*)carve((size_t)n_drug * sizeof(float));
  const size_t zero_bytes = off;  // accumulators must start at 0
  float* h_prot = (float*)carve((size_t)n_prot * HFEAT * sizeof(float));
  float* h_drug = (float*)carve((size_t)n_drug * HFEAT * sizeof(float));
  (void)ws_size;

  hipMemsetAsync(d_ws, 0, zero_bytes, stream);  // capturable memset node

  // Phase 1: scatter-mean numerators + counts (both edge directions fused).
  {
    const int total = E * (DFEAT / 4);
    hgnn_scatter_kernel<<<(total + 255) / 256, 256, 0, stream>>>(
        x_drug, x_prot, src_drug, dst_prot, sum_p, cnt_p, sum_d, cnt_d, E);
  }

  // Phase 2: two SAGE GEMMs with fp32 WMMA (8 waves per 16-row stripe).
  {
    hgnn_sage_gemm_wmma<<<(n_prot + 15) / 16, 256, 0, stream>>>(
        sum_p, cnt_p, x_prot, Wl_dp, bl_dp, Wr_dp, h_prot, n_prot);
  }
  {
    hgnn_sage_gemm_wmma<<<(n_drug + 15) / 16, 256, 0, stream>>>(
        sum_d, cnt_d, x_drug, Wl_pd, bl_pd, Wr_pd, h_drug, n_drug);
  }

  // Phase 3: edge classifier (wave-per-edge, W_lin in LDS).
  {
    const int wavesPerBlock = 8;   // 256 threads = 8 wave32
    const int blocks = (L + wavesPerBlock - 1) / wavesPerBlock;
    hgnn_edge_mlp_kernel<<<blocks, 32 * wavesPerBlock, 0, stream>>>(
        h_drug, h_prot, el_drug, el_prot, W_lin, b_lin, (float*)d_out, L);
  }
}